// GAT_6270652252998
// MI455X (gfx1250) — compile-verified
//
#include <hip/hip_runtime.h>
#include <hip/hip_bf16.h>
#include <math.h>

// ---------------------------------------------------------------------------
// HAN forward (2 layers) for MI455X / gfx1250, wave32, WMMA f32 16x16x4.
// GEMMs keep full W in LDS (<=64KB); edge phase is atomics/memory bound.
// ---------------------------------------------------------------------------

typedef __attribute__((ext_vector_type(2))) float v2f;
typedef __attribute__((ext_vector_type(8))) float v8f;

#define NN 100000
#define EE 500000

// order-preserving float <-> uint for atomic max
__device__ __forceinline__ unsigned fenc(float f) {
  unsigned u = __float_as_uint(f);
  return (u & 0x80000000u) ? ~u : (u | 0x80000000u);
}
__device__ __forceinline__ float fdec(unsigned u) {
  u = (u & 0x80000000u) ? (u & 0x7fffffffu) : ~u;
  return __uint_as_float(u);
}

// ---------------------------------------------------------------------------
// WMMA fp32 GEMM: C[n,Cout] = A[n,Cin] @ W[Cin,Cout] + bias,  Cout = NT*16.
//   mode 0: store C
//   mode 1: kacc[col] += invN * sum_rows tanh(C)   (semantic-attention kfeat)
// Block = 256 threads (8 waves). Whole W is staged in LDS; each wave computes
// a 16-row band across all NT column tiles, so each A fragment feeds NT
// back-to-back WMMAs (independent accumulators -> no D->A/B hazards).
// n % 16 == 0, Cin % 4 == 0 (true for this model: 128/64).
// ---------------------------------------------------------------------------
template <int NT>
__global__ void wmma_gemm_bias(const float* __restrict__ A,
                               const float* __restrict__ W,
                               const float* __restrict__ bias,
                               float* __restrict__ C,
                               float* __restrict__ kacc,
                               int n, int Cin, int mode, float invN) {
  constexpr int Cout = NT * 16;
  __shared__ float Bs[128 * Cout];  // Cin <= 128

  const int wave = threadIdx.x >> 5;
  const int lane = threadIdx.x & 31;
  const int tm = blockIdx.x * 128 + wave * 16;

  // cooperative copy of the whole weight matrix into LDS (coalesced b128)
  {
    const float4* W4 = (const float4*)W;
    float4* B4 = (float4*)Bs;
    const int n4 = (Cin * Cout) >> 2;
    for (int idx = threadIdx.x; idx < n4; idx += 256) B4[idx] = W4[idx];
  }
  __syncthreads();

  if (tm >= n) return;  // wave-uniform, after the barrier

  const int mlo = lane & 15;  // A row / B,C col within tile
  const int hi = lane >> 4;   // half-select

  v8f zero = {0.f, 0.f, 0.f, 0.f, 0.f, 0.f, 0.f, 0.f};
  v8f acc[NT];
#pragma unroll
  for (int t = 0; t < NT; ++t) acc[t] = zero;

  // A frag: lanes 0-15 -> K={k,k+1}; lanes 16-31 -> K={k+2,k+3}, row = mlo
  const float* Arow = A + (size_t)(tm + mlo) * Cin + 2 * hi;
  __builtin_prefetch(Arow + 64, 0, 1);  // global_prefetch_b8: warm A stream

  for (int k = 0; k < Cin; k += 4) {
    v2f a;
    a.x = Arow[k];
    a.y = Arow[k + 1];
    const float* brow = Bs + (k + 2 * hi) * Cout + mlo;
#pragma unroll
    for (int t = 0; t < NT; ++t) {
      v2f b;
      b.x = brow[t * 16];         // row k+2*hi
      b.y = brow[t * 16 + Cout];  // row k+2*hi+1
      acc[t] = __builtin_amdgcn_wmma_f32_16x16x4_f32(false, a, false, b,
                                                     (short)0, acc[t],
                                                     false, false);
    }
  }

  if (mode == 0) {
#pragma unroll
    for (int t = 0; t < NT; ++t) {
      const int col = t * 16 + mlo;
      const float bv = bias[col];
#pragma unroll
      for (int r = 0; r < 8; ++r) {
        int row = tm + r + 8 * hi;
        C[(size_t)row * Cout + col] = acc[t][r] + bv;
      }
    }
  } else {
#pragma unroll
    for (int t = 0; t < NT; ++t) {
      const int col = t * 16 + mlo;
      const float bv = bias[col];
      float s = 0.f;
#pragma unroll
      for (int r = 0; r < 8; ++r) s += tanhf(acc[t][r] + bv);
      atomicAdd(&kacc[col], s * invN);
    }
  }
}

// out[n*H+h] = sum_d h[n, h, d] * avec[h, d]   (D multiple of 4)
__global__ void attn_logits(const float* __restrict__ h,
                            const float* __restrict__ avec,
                            float* __restrict__ out, int n, int H, int D) {
  int i = blockIdx.x * blockDim.x + threadIdx.x;
  if (i >= n * H) return;
  int node = i / H, hh = i - node * H;
  const float4* p = (const float4*)(h + (size_t)node * H * D + (size_t)hh * D);
  const float4* a = (const float4*)(avec + (size_t)hh * D);
  float s = 0.f;
  for (int d = 0; d < (D >> 2); ++d) {
    float4 x = p[d], y = a[d];
    s += x.x * y.x + x.y * y.y + x.z * y.z + x.w * y.w;
  }
  out[i] = s;
}

// pass1 (per edge): logit = leaky_relu(a_s[src]+a_d[dst]); segment max
__global__ void edge_pass1(const int* __restrict__ src, const int* __restrict__ dst,
                           const float* __restrict__ as_, const float* __restrict__ ad_,
                           float* __restrict__ lg, unsigned* __restrict__ smax,
                           int E, int H) {
  int e = blockIdx.x * blockDim.x + threadIdx.x;
  if (e >= E) return;
  int s = src[e], d = dst[e];
  const float* pa = as_ + (size_t)s * H;
  const float* pd = ad_ + (size_t)d * H;
  float* lge = lg + (size_t)e * H;
  unsigned* sm = smax + (size_t)d * H;
  for (int h = 0; h < H; ++h) {
    float l = pa[h] + pd[h];
    l = (l > 0.f) ? l : 0.2f * l;
    lge[h] = l;
    atomicMax(&sm[h], fenc(l));
  }
}

// pass2 (per edge): ex = exp(logit - max[dst]); segment sum
__global__ void edge_pass2(const int* __restrict__ dst,
                           float* __restrict__ lg, const unsigned* __restrict__ smax,
                           float* __restrict__ ssum, int E, int H) {
  int e = blockIdx.x * blockDim.x + threadIdx.x;
  if (e >= E) return;
  int d = dst[e];
  float* lge = lg + (size_t)e * H;
  const unsigned* sm = smax + (size_t)d * H;
  float* ss = ssum + (size_t)d * H;
  for (int h = 0; h < H; ++h) {
    float m = fdec(sm[h]);
    float ex = expf(lge[h] - m);
    lge[h] = ex;
    atomicAdd(&ss[h], ex);
  }
}

// pass3: agg[dst, f..f+3] += h_src[src, f..f+3] * ex/(sum[dst,h]+1e-16)
__global__ void edge_pass3(const int* __restrict__ src, const int* __restrict__ dst,
                           const float* __restrict__ hsrc, const float* __restrict__ lg,
                           const float* __restrict__ ssum, float* __restrict__ agg,
                           int E, int H, int F, int logD) {
  const int F4 = F >> 2;
  long long i = (long long)blockIdx.x * blockDim.x + threadIdx.x;
  if (i >= (long long)E * F4) return;
  int e = (int)(i / F4);
  int f = (int)(i - (long long)e * F4) << 2;
  int hh = f >> logD;
  int s = src[e], d = dst[e];
  float w = lg[(size_t)e * H + hh] / (ssum[(size_t)d * H + hh] + 1e-16f);
  float4 hv = *(const float4*)(hsrc + (size_t)s * F + f);
  float* ap = agg + (size_t)d * F + f;
  atomicAdd(ap + 0, hv.x * w);
  atomicAdd(ap + 1, hv.y * w);
  atomicAdd(ap + 2, hv.z * w);
  atomicAdd(ap + 3, hv.w * w);
}

__global__ void relu_k(float* __restrict__ p, long long n4) {
  long long i = (long long)blockIdx.x * blockDim.x + threadIdx.x;
  if (i >= n4) return;
  float4* p4 = (float4*)p;
  float4 v = p4[i];
  v.x = fmaxf(v.x, 0.f); v.y = fmaxf(v.y, 0.f);
  v.z = fmaxf(v.z, 0.f); v.w = fmaxf(v.w, 0.f);
  p4[i] = v;
}

// attn = softmax over K=2 of sum_f q[f] * kfeat[k, f]
__global__ void sem_attn(const float* __restrict__ kfeat, const float* __restrict__ q,
                         float* __restrict__ attn, int F) {
  if (threadIdx.x == 0 && blockIdx.x == 0) {
    float s0 = 0.f, s1 = 0.f;
    for (int f = 0; f < F; ++f) { s0 += q[f] * kfeat[f]; s1 += q[f] * kfeat[F + f]; }
    float m = fmaxf(s0, s1);
    float e0 = expf(s0 - m), e1 = expf(s1 - m);
    float inv = 1.f / (e0 + e1);
    attn[0] = e0 * inv;
    attn[1] = e1 * inv;
  }
}

__device__ __forceinline__ float elu1(float v) {
  return (v > 0.f) ? v : (expf(v) - 1.f);
}

// res = attn0*o0 + attn1*o1  (+ optional ELU for inter-layer activation)
__global__ void combine_k(const float* __restrict__ o0, const float* __restrict__ o1,
                          const float* __restrict__ attn, float* __restrict__ res,
                          long long n4, int do_elu) {
  long long i = (long long)blockIdx.x * blockDim.x + threadIdx.x;
  if (i >= n4) return;
  float a0 = attn[0], a1 = attn[1];
  float4 x = ((const float4*)o0)[i];
  float4 y = ((const float4*)o1)[i];
  float4 v;
  v.x = a0 * x.x + a1 * y.x;
  v.y = a0 * x.y + a1 * y.y;
  v.z = a0 * x.z + a1 * y.z;
  v.w = a0 * x.w + a1 * y.w;
  if (do_elu) { v.x = elu1(v.x); v.y = elu1(v.y); v.z = elu1(v.z); v.w = elu1(v.w); }
  ((float4*)res)[i] = v;
}

// ---------------------------------------------------------------------------
extern "C" void kernel_launch(void* const* d_in, const int* in_sizes, int n_in,
                              void* d_out, int out_size, void* d_ws, size_t ws_size,
                              hipStream_t stream) {
  const int N = NN, E = EE;

  const float* x_p = (const float*)d_in[0];
  const float* x_a = (const float*)d_in[1];
  // EDGE_META order: pp, aa, ap, pa  (== d_in[2..5])
  const int* ei[4] = {(const int*)d_in[2], (const int*)d_in[3],
                      (const int*)d_in[4], (const int*)d_in[5]};
  static const int stv[4] = {0, 1, 1, 0};     // src type: 0=paper 1=author
  static const int dtv[4] = {0, 1, 0, 1};     // dst type
  static const int aggmap[4] = {0, 2, 1, 3};  // pp->p0, aa->a0, ap->p1, pa->a1

  void* const* P1 = d_in + 6;   // W_p,b_p,W_a,b_a,kW,kb,q, (asrc,adst)x4
  void* const* P2 = d_in + 21;

  // ---- workspace layout (floats) ----
  float* w = (float*)d_ws;
  const size_t NF1 = (size_t)N * 128;
  float* h_nt[2] = {w, w + NF1};                       // projected features
  float* agg[4] = {w + 2 * NF1, w + 3 * NF1,           // [0]=paper0 [1]=paper1
                   w + 4 * NF1, w + 5 * NF1};          // [2]=author0 [3]=author1
  float* x1_p = w + 6 * NF1;
  float* x1_a = w + 7 * NF1;
  float* alpha = w + 8 * NF1;                          // E*H
  float* asb = alpha + (size_t)E * 8;                  // N*H
  float* adb = asb + (size_t)N * 8;                    // N*H
  unsigned* smax = (unsigned*)(adb + (size_t)N * 8);   // N*H
  float* ssum = (float*)smax + (size_t)N * 8;          // N*H
  float* kfeat = ssum + (size_t)N * 8;                 // 2*F
  float* attn = kfeat + 256;                           // 2

  const int TB = 256;
  const int gemm_grid = (N + 127) / 128;

  auto launch_gemm = [&](const float* A, const float* W, const float* b,
                         float* C, float* kacc, int Cin, int Cout,
                         int mode, float invN) {
    if (Cout == 128)
      wmma_gemm_bias<8><<<gemm_grid, 256, 0, stream>>>(A, W, b, C, kacc,
                                                       N, Cin, mode, invN);
    else
      wmma_gemm_bias<4><<<gemm_grid, 256, 0, stream>>>(A, W, b, C, kacc,
                                                       N, Cin, mode, invN);
  };

  auto layer = [&](const float* xp, const float* xa, void* const* PP,
                   int Fin, int Fout, int H, int D, int logD,
                   float* out_p, float* out_a, int do_elu) {
    const float* Wp = (const float*)PP[0]; const float* bp = (const float*)PP[1];
    const float* Wa = (const float*)PP[2]; const float* ba = (const float*)PP[3];
    const float* kW = (const float*)PP[4]; const float* kb = (const float*)PP[5];
    const float* qv = (const float*)PP[6];

    // node projections: h = x @ W + b
    launch_gemm(xp, Wp, bp, h_nt[0], nullptr, Fin, Fout, 0, 0.f);
    launch_gemm(xa, Wa, ba, h_nt[1], nullptr, Fin, Fout, 0, 0.f);

    for (int k = 0; k < 4; ++k)
      hipMemsetAsync(agg[k], 0, (size_t)N * Fout * sizeof(float), stream);

    const size_t nh = (size_t)N * H;

    for (int et = 0; et < 4; ++et) {
      const float* hs = h_nt[stv[et]];
      const float* hd = h_nt[dtv[et]];
      const float* asrc = (const float*)PP[7 + 2 * et];
      const float* adst = (const float*)PP[8 + 2 * et];
      float* aggo = agg[aggmap[et]];
      const int* s = ei[et];
      const int* d = ei[et] + E;

      attn_logits<<<(int)((nh + TB - 1) / TB), TB, 0, stream>>>(hs, asrc, asb,
                                                                N, H, D);
      attn_logits<<<(int)((nh + TB - 1) / TB), TB, 0, stream>>>(hd, adst, adb,
                                                                N, H, D);

      hipMemsetAsync(smax, 0, nh * sizeof(unsigned), stream);
      hipMemsetAsync(ssum, 0, nh * sizeof(float), stream);

      edge_pass1<<<(E + TB - 1) / TB, TB, 0, stream>>>(s, d, asb, adb, alpha,
                                                       smax, E, H);
      edge_pass2<<<(E + TB - 1) / TB, TB, 0, stream>>>(d, alpha, smax, ssum,
                                                       E, H);
      long long ef4 = (long long)E * (Fout >> 2);
      edge_pass3<<<(int)((ef4 + TB - 1) / TB), TB, 0, stream>>>(
          s, d, hs, alpha, ssum, aggo, E, H, Fout, logD);
    }

    long long nf4 = (long long)N * (Fout >> 2);
    for (int k = 0; k < 4; ++k)
      relu_k<<<(int)((nf4 + TB - 1) / TB), TB, 0, stream>>>(agg[k], nf4);

    // semantic attention per destination node type
    for (int nt = 0; nt < 2; ++nt) {
      float* o0 = agg[2 * nt];      // paper: {pp, ap}; author: {aa, pa}
      float* o1 = agg[2 * nt + 1];
      hipMemsetAsync(kfeat, 0, 2 * Fout * sizeof(float), stream);
      launch_gemm(o0, kW, kb, nullptr, kfeat, Fout, Fout, 1, 1.f / N);
      launch_gemm(o1, kW, kb, nullptr, kfeat + Fout, Fout, Fout, 1, 1.f / N);
      sem_attn<<<1, 1, 0, stream>>>(kfeat, qv, attn, Fout);
      float* res = (nt == 0) ? out_p : out_a;
      combine_k<<<(int)((nf4 + TB - 1) / TB), TB, 0, stream>>>(o0, o1, attn, res,
                                                               nf4, do_elu);
    }
  };

  // layer 1: 128 -> 128, 8 heads of dim 16, ELU on output
  layer(x_p, x_a, P1, 128, 128, 8, 16, 4, x1_p, x1_a, 1);
  // layer 2: 128 -> 64, 1 head of dim 64, write straight into d_out
  float* out_p = (float*)d_out;
  float* out_a = (float*)d_out + (size_t)N * 64;
  layer(x1_p, x1_a, P2, 128, 64, 1, 64, 6, out_p, out_a, 0);
}